// EnhancedGrayscaleCombinedLoss_21019569947115
// MI455X (gfx1250) — compile-verified
//
#include <hip/hip_runtime.h>

typedef float v2f __attribute__((ext_vector_type(2)));
typedef float v8f __attribute__((ext_vector_type(8)));

#define C1F 1.0e-4f
#define C2F 9.0e-4f

__device__ __forceinline__ int clampi(int v, int lo, int hi) {
  return v < lo ? lo : (v > hi ? hi : v);
}

// ---------------------------------------------------------------------------
// Fused per-scale SSIM kernel. One wave (32 threads) computes TPW consecutive
// 16x16 output tiles via fp32 WMMA: separable Gaussian conv of 5 maps
// (x, y, xx, yy, xy) as banded GEMMs, then the l*cs / cs pointwise math.
// Banded-weight VGPRs and Gaussian LDS tables are built once and reused for
// all TPW tiles; one fp64 slot-atomic pair per wave.
// ---------------------------------------------------------------------------
template <int KH, int KW, int TPW>
__global__ __launch_bounds__(32) void ssim_scale(
    const float* __restrict__ X, const float* __restrict__ Y,
    int H, int W, int OH, int OW, int TW,
    double* __restrict__ lcs_slots, double* __restrict__ cs_slots) {
  constexpr int PAD_H = KH / 2, PAD_W = KW / 2;
  constexpr int EXT_W = 16 + KW - 1;         // horizontal extended width
  constexpr int EXT_H = 16 + KH - 1;         // vertical extended height
  constexpr int CH_H  = (EXT_W + 3) / 4;     // K-chunks, horizontal pass
  constexpr int CH_V  = (EXT_H + 3) / 4;     // K-chunks, vertical pass

  __shared__ float wH[KW];
  __shared__ float wV[KH];
  __shared__ float Tbuf[32][17];             // horizontal result staging

  const int lane = threadIdx.x;              // 0..31, wave32
  const int n16  = lane & 15;
  const int kofs = (lane < 16) ? 0 : 2;      // A/B K-pair selector (ISA layout)
  const int rofs = (lane < 16) ? 0 : 8;      // C/D row offset (ISA layout)

  // --- Gaussian weights (match reference _gauss_1d, incl. even-size center) ---
  if (lane == 0) {
    const float sh = 1.5f;
    const float sw = sh * (float)KW / (float)KH;
    float tmp[(KW > KH) ? KW : KH];
    float s = 0.f;
    for (int i = 0; i < KH; ++i) {
      float d = (float)(i - KH / 2);
      tmp[i] = expf(-(d * d) / (2.f * sh * sh));
      s += tmp[i];
    }
    for (int i = 0; i < KH; ++i) wV[i] = tmp[i] / s;
    s = 0.f;
    for (int i = 0; i < KW; ++i) {
      float d = (float)(i - KW / 2);
      tmp[i] = expf(-(d * d) / (2.f * sw * sw));
      s += tmp[i];
    }
    for (int i = 0; i < KW; ++i) wH[i] = tmp[i] / s;
  }
  __syncthreads();

  // --- Banded weight matrices: lane-dependent only, reused for all tiles ---
  // Horizontal B chunk (4x16): B[kk][n] = wH[(4k+kk) - n] if in band else 0.
  v2f bh[CH_H];
#pragma unroll
  for (int k = 0; k < CH_H; ++k) {
    int d0 = 4 * k + kofs - n16, d1 = d0 + 1;
    float v0 = wH[clampi(d0, 0, KW - 1)];    // unconditional LDS load
    float v1 = wH[clampi(d1, 0, KW - 1)];
    bh[k].x = ((unsigned)d0 < (unsigned)KW) ? v0 : 0.f;  // select, no branch
    bh[k].y = ((unsigned)d1 < (unsigned)KW) ? v1 : 0.f;
  }
  // Vertical A chunk (16x4): A[m][kk] = wV[(4k+kk) - m] if in band else 0.
  v2f av[CH_V];
#pragma unroll
  for (int k = 0; k < CH_V; ++k) {
    int d0 = 4 * k + kofs - n16, d1 = d0 + 1;
    float v0 = wV[clampi(d0, 0, KH - 1)];
    float v1 = wV[clampi(d1, 0, KH - 1)];
    av[k].x = ((unsigned)d0 < (unsigned)KH) ? v0 : 0.f;
    av[k].y = ((unsigned)d1 < (unsigned)KH) ? v1 : 0.f;
  }

  const int img = blockIdx.z;
  const int tr  = blockIdx.y;
  const int rb  = tr * 16 - PAD_H;           // first extended input row
  const float* Xi = X + (size_t)img * H * (size_t)W;
  const float* Yi = Y + (size_t)img * H * (size_t)W;

  float sum_lcs = 0.f, sum_cs = 0.f;

  for (int t = 0; t < TPW; ++t) {
    const int tc = blockIdx.x * TPW + t;     // wave-uniform
    if (tc >= TW) break;                     // uniform branch: EXEC stays all-1
    const int cb = tc * 16 - PAD_W;          // first extended input col

    // --- Load x/y A-matrix elements for both row-halves (zero-padded) ---
    // A layout (16x4 f32): lane holds row M=lane&15; K = 4k + kofs + {0,1}.
    float xv[2][2 * CH_H], yv[2][2 * CH_H];
#pragma unroll
    for (int h = 0; h < 2; ++h) {
      int grow = rb + h * 16 + n16;
      bool rok = (unsigned)grow < (unsigned)H;
      const float* xr = Xi + (size_t)clampi(grow, 0, H - 1) * (size_t)W;
      const float* yr = Yi + (size_t)clampi(grow, 0, H - 1) * (size_t)W;
#pragma unroll
      for (int k = 0; k < CH_H; ++k) {
        int c0 = cb + 4 * k + kofs;
        int c1 = c0 + 1;
        bool ok0 = rok && ((unsigned)c0 < (unsigned)W);
        bool ok1 = rok && ((unsigned)c1 < (unsigned)W);
        float xa = xr[clampi(c0, 0, W - 1)];   // unconditional loads
        float xb = xr[clampi(c1, 0, W - 1)];
        float ya = yr[clampi(c0, 0, W - 1)];
        float yb = yr[clampi(c1, 0, W - 1)];
        xv[h][2 * k]     = ok0 ? xa : 0.f;     // selects, no exec branching
        xv[h][2 * k + 1] = ok1 ? xb : 0.f;
        yv[h][2 * k]     = ok0 ? ya : 0.f;
        yv[h][2 * k + 1] = ok1 ? yb : 0.f;
      }
    }

    v8f res[5];
#pragma unroll
    for (int p = 0; p < 5; ++p) {
      // --- Horizontal pass: two 16-row halves of the extended tile ---
      v8f th[2];
#pragma unroll
      for (int h = 0; h < 2; ++h) {
        v8f acc = {};
#pragma unroll
        for (int k = 0; k < CH_H; ++k) {
          float x0 = xv[h][2 * k], x1 = xv[h][2 * k + 1];
          float y0 = yv[h][2 * k], y1 = yv[h][2 * k + 1];
          float a0, a1;
          if (p == 0)      { a0 = x0;      a1 = x1; }
          else if (p == 1) { a0 = y0;      a1 = y1; }
          else if (p == 2) { a0 = x0 * x0; a1 = x1 * x1; }
          else if (p == 3) { a0 = y0 * y0; a1 = y1 * y1; }
          else             { a0 = x0 * y0; a1 = x1 * y1; }
          v2f a = {a0, a1};
          acc = __builtin_amdgcn_wmma_f32_16x16x4_f32(
              false, a, false, bh[k], (short)0, acc, false, false);
        }
        th[h] = acc;
      }
      // --- Stage horizontal result in LDS (C/D layout -> row-major) ---
      __syncthreads();  // previous vertical reads complete before overwrite
#pragma unroll
      for (int j = 0; j < 8; ++j) {
        Tbuf[rofs + j][n16]      = th[0][j];
        Tbuf[16 + rofs + j][n16] = th[1][j];
      }
      __syncthreads();
      // --- Vertical pass: banded Wv(16x4) x T(4x16) chunks ---
      v8f vac = {};
#pragma unroll
      for (int k = 0; k < CH_V; ++k) {
        v2f b = {Tbuf[4 * k + kofs][n16], Tbuf[4 * k + kofs + 1][n16]};
        vac = __builtin_amdgcn_wmma_f32_16x16x4_f32(
            false, av[k], false, b, (short)0, vac, false, false);
      }
      res[p] = vac;
    }

    // --- Pointwise SSIM math + masked accumulation over this tile ---
#pragma unroll
    for (int j = 0; j < 8; ++j) {
      int orow = tr * 16 + rofs + j;
      int ocol = tc * 16 + n16;
      float mux = res[0][j], muy = res[1][j];
      float sxx = res[2][j] - mux * mux;
      float syy = res[3][j] - muy * muy;
      float sxy = res[4][j] - mux * muy;
      float l  = (2.f * mux * muy + C1F) / (mux * mux + muy * muy + C1F);
      float cs = (2.f * sxy + C2F) / (sxx + syy + C2F);
      bool ok = (orow < OH) && (ocol < OW);
      sum_lcs += ok ? l * cs : 0.f;
      sum_cs  += ok ? cs : 0.f;
    }
  }

#pragma unroll
  for (int off = 16; off >= 1; off >>= 1) {
    sum_lcs += __shfl_xor(sum_lcs, off, 32);
    sum_cs  += __shfl_xor(sum_cs, off, 32);
  }
  if (lane == 0) {
    int slot = (int)((blockIdx.x +
                      gridDim.x * (blockIdx.y + gridDim.y * blockIdx.z)) & 255);
    atomicAdd(&lcs_slots[slot], (double)sum_lcs);
    atomicAdd(&cs_slots[slot], (double)sum_cs);
  }
}

// ---------------------------------------------------------------------------
// 2x2 average pool (scale chain).
// ---------------------------------------------------------------------------
__global__ void avgpool2_k(const float* __restrict__ in, float* __restrict__ out,
                           int N, int H, int W) {
  int HO = H >> 1, WO = W >> 1;
  size_t total = (size_t)N * HO * WO;
  for (size_t i = (size_t)blockIdx.x * blockDim.x + threadIdx.x; i < total;
       i += (size_t)gridDim.x * blockDim.x) {
    int j = (int)(i % WO);
    size_t t = i / WO;
    int r = (int)(t % HO);
    int n = (int)(t / HO);
    const float* p = in + ((size_t)n * H + 2 * r) * (size_t)W + 2 * j;
    out[i] = 0.25f * (p[0] + p[1] + p[W] + p[W + 1]);
  }
}

// ---------------------------------------------------------------------------
// L2 partial reduction (grid-stride), fp64 slot atomics.
// ---------------------------------------------------------------------------
__global__ void l2_k(const float* __restrict__ X, const float* __restrict__ Y,
                     size_t total, double* __restrict__ slots) {
  float s = 0.f;
  for (size_t i = (size_t)blockIdx.x * blockDim.x + threadIdx.x; i < total;
       i += (size_t)gridDim.x * blockDim.x) {
    float d = X[i] - Y[i];
    s += d * d;
  }
#pragma unroll
  for (int off = 16; off >= 1; off >>= 1) s += __shfl_xor(s, off, 32);
  __shared__ float ls[8];
  int lane = threadIdx.x & 31, w = threadIdx.x >> 5;
  if (lane == 0) ls[w] = s;
  __syncthreads();
  if (threadIdx.x == 0) {
    float t = 0.f;
    for (int i = 0; i < 8; ++i) t += ls[i];
    atomicAdd(&slots[blockIdx.x & 255], (double)t);
  }
}

// ---------------------------------------------------------------------------
// Final combine: means, weighted power product, loss stack.
// ---------------------------------------------------------------------------
__global__ void finalize_k(const double* __restrict__ acc, float* __restrict__ out) {
  if (threadIdx.x != 0 || blockIdx.x != 0) return;
  double s[7];
  for (int g = 0; g < 7; ++g) {
    double a = 0.0;
    for (int i = 0; i < 256; ++i) a += acc[g * 256 + i];
    s[g] = a;
  }
  // group order: lcs0, cs0, lcs1, cs1, lcs2, cs2, l2
  const double c0 = 64.0 * 176.0 * 1008.0;   // scale0 map elems (also l2 count)
  const double c1 = 64.0 * 88.0 * 505.0;     // scale1 map elems (out_w = W+1)
  const double c2 = 64.0 * 44.0 * 253.0;     // scale2 map elems
  double m_cs0  = s[1] / c0;
  double m_cs1  = s[3] / c1;
  double m_lcs2 = s[4] / c2;
  const double wsum = 0.0448 + 0.2856 + 0.3001;
  double ms = pow(m_cs0, 0.0448 / wsum) * pow(m_cs1, 0.2856 / wsum) *
              pow(m_lcs2, 0.3001 / wsum);
  double l2m = s[6] / c0;
  double sl  = 1.0 - ms;
  double tot = 0.3 * l2m + 0.7 * sl;
  out[0] = (float)tot;
  out[1] = (float)l2m;
  out[2] = (float)sl;
}

// ---------------------------------------------------------------------------
extern "C" void kernel_launch(void* const* d_in, const int* in_sizes, int n_in,
                              void* d_out, int out_size, void* d_ws, size_t ws_size,
                              hipStream_t stream) {
  (void)in_sizes; (void)n_in; (void)out_size; (void)ws_size;
  const float* X = (const float*)d_in[0];
  const float* Y = (const float*)d_in[1];
  float* out = (float*)d_out;

  // ws layout: [0, 1792) doubles = accumulator slots; then pyramid buffers.
  double* acc = (double*)d_ws;
  float* x1 = (float*)((char*)d_ws + 16384);
  const size_t n1 = (size_t)64 * 88 * 504;   // 2,838,528
  const size_t n2 = (size_t)64 * 44 * 252;   //   709,632
  float* y1 = x1 + n1;
  float* x2 = y1 + n1;
  float* y2 = x2 + n2;

  hipMemsetAsync(d_ws, 0, 1792 * sizeof(double), stream);

  // Scale pyramid (avg-pool 2x2).
  int blocks1 = (int)((n1 + 255) / 256);
  avgpool2_k<<<blocks1, 256, 0, stream>>>(X, x1, 64, 176, 1008);
  avgpool2_k<<<blocks1, 256, 0, stream>>>(Y, y1, 64, 176, 1008);
  int blocks2 = (int)((n2 + 255) / 256);
  avgpool2_k<<<blocks2, 256, 0, stream>>>(x1, x2, 64, 88, 504);
  avgpool2_k<<<blocks2, 256, 0, stream>>>(y1, y2, 64, 88, 504);

  // Fused SSIM per scale: grid = (ceil(TW/4), tiles_h, N), one wave per block,
  // 4 row-adjacent tiles per wave (weights/bands amortized).
  ssim_scale<11, 15, 4><<<dim3(16, 11, 64), 32, 0, stream>>>(
      X, Y, 176, 1008, 176, 1008, 63, acc + 0, acc + 256);
  ssim_scale<9, 12, 4><<<dim3(8, 6, 64), 32, 0, stream>>>(
      x1, y1, 88, 504, 88, 505, 32, acc + 512, acc + 768);
  ssim_scale<7, 10, 4><<<dim3(4, 3, 64), 32, 0, stream>>>(
      x2, y2, 44, 252, 44, 253, 16, acc + 1024, acc + 1280);

  // L2 over full-resolution inputs (L2-cache resident after ssim pass).
  l2_k<<<2048, 256, 0, stream>>>(X, Y, (size_t)64 * 176 * 1008, acc + 1536);

  finalize_k<<<1, 1, 0, stream>>>(acc, out);
}